// CBOWModel_32985348833309
// MI455X (gfx1250) — compile-verified
//
#include <hip/hip_runtime.h>
#include <math.h>

#define VOCAB 100000
#define DIM   128
#define BATCH 16384
#define CTX   10
#define NNEG  10
#define ROWPAD 132   // 128 + 4 pad floats -> conflict-free stride-132 LDS reads

typedef __attribute__((ext_vector_type(2))) float v2f;
typedef __attribute__((ext_vector_type(8))) float v8f;

// One block = 16 batch elements. 4 waves of 32 (wave32).
__global__ __launch_bounds__(128) void cbow_tile_kernel(
    const int*   __restrict__ pos_u,     // [B, CTX]
    const int*   __restrict__ pos_v,     // [B]
    const int*   __restrict__ neg_v,     // [B, NNEG]
    const float* __restrict__ u_weight,  // [VOCAB, DIM]
    const float* __restrict__ v_weight,  // [VOCAB, DIM]
    float*       __restrict__ partials)  // [gridDim.x]
{
    __shared__ float Us[16 * ROWPAD];          //  8.4 KB
    __shared__ float Vs[4 * 16 * ROWPAD];      // 33.8 KB (4 j-slots at a time)
    __shared__ float red[128];

    const int tid  = threadIdx.x;
    const int lane = tid & 31;
    const int wave = tid >> 5;
    const int b0   = blockIdx.x * 16;

    // ---- Phase 1: U = mean over CTX of gathered u_weight rows ----
    {
        const int m = tid >> 3;         // 0..15 : batch row in tile
        const int f = (tid & 7) * 16;   // 16-float chunk within the row
        float acc[16];
        #pragma unroll
        for (int k = 0; k < 16; ++k) acc[k] = 0.f;
        for (int c = 0; c < CTX; ++c) {
            const int idx = pos_u[(b0 + m) * CTX + c];
            const float4* src = (const float4*)(u_weight + (long)idx * DIM + f);
            #pragma unroll
            for (int q = 0; q < 4; ++q) {
                float4 x = src[q];
                acc[q * 4 + 0] += x.x; acc[q * 4 + 1] += x.y;
                acc[q * 4 + 2] += x.z; acc[q * 4 + 3] += x.w;
            }
        }
        #pragma unroll
        for (int k = 0; k < 16; ++k)
            Us[m * ROWPAD + f + k] = acc[k] * 0.1f;   // /CTX
    }

    float loss = 0.f;   // per-lane accumulated loss (only diag-holding lanes)

    // ---- Phase 2: 11 dot-product slots (j=0 pos, j=1..10 neg), 4 per pass ----
    for (int chunkBase = 0; chunkBase < 11; chunkBase += 4) {
        __syncthreads();
        {   // load up to 4 v-slots into LDS with coalesced b128 gathers
            const int m = tid >> 3;
            const int f = (tid & 7) * 16;
            for (int p = 0; p < 4; ++p) {
                const int j = chunkBase + p;
                if (j >= 11) break;
                const int bi  = b0 + m;
                const int idx = (j == 0) ? pos_v[bi] : neg_v[bi * NNEG + (j - 1)];
                const float4* src = (const float4*)(v_weight + (long)idx * DIM + f);
                float* dst = Vs + (p * 16 + m) * ROWPAD + f;
                #pragma unroll
                for (int q = 0; q < 4; ++q)
                    ((float4*)dst)[q] = src[q];
            }
        }
        __syncthreads();

        const int j = chunkBase + wave;   // wave-uniform -> scalar branch, EXEC stays full
        if (j < 11) {
            // C(16x16) = U(16x128) x V_j^T(128x16) via 32-step f32 WMMA chain.
            // A 16x4 frag: lanes 0-15 row=lane K={4t,4t+1}; lanes 16-31 row=lane-16 K={4t+2,4t+3}
            // B 4x16 frag: identical per-lane addressing into V_j rows.
            const int row  = lane & 15;
            const int koff = (lane >> 4) * 2;
            const float* ua = Us + row * ROWPAD + koff;
            const float* vb = Vs + (wave * 16 + row) * ROWPAD + koff;
            v8f c = {};
            #pragma unroll
            for (int t = 0; t < DIM / 4; ++t) {
                v2f a = *(const v2f*)(ua + 4 * t);
                v2f b = *(const v2f*)(vb + 4 * t);
                c = __builtin_amdgcn_wmma_f32_16x16x4_f32(
                        false, a, false, b, (short)0, c, false, false);
            }
            // Diagonal: (m,m) lives at VGPR r, lane r (m=r) and VGPR r, lane 24+r (m=8+r)
            float s = 0.f;
            bool  has = false;
            #pragma unroll
            for (int r = 0; r < 8; ++r) {
                if (lane == r || lane == 24 + r) { s = c[r]; has = true; }
            }
            if (has) {
                s = fminf(fmaxf(s, -10.f), 10.f);
                // j==0: -log_sigmoid(s) = log(1+e^-s);  j>0: -log_sigmoid(-s) = log(1+e^s)
                loss += (j == 0) ? logf(1.f + expf(-s)) : logf(1.f + expf(s));
            }
        }
    }

    // ---- Phase 3: deterministic block reduction ----
    __syncthreads();
    red[tid] = loss;
    __syncthreads();
    #pragma unroll
    for (int off = 64; off > 0; off >>= 1) {
        if (tid < off) red[tid] += red[tid + off];
        __syncthreads();
    }
    if (tid == 0) partials[blockIdx.x] = red[0];
}

__global__ __launch_bounds__(256) void cbow_reduce_kernel(
    const float* __restrict__ partials, int n, float* __restrict__ out)
{
    __shared__ float red[256];
    float s = 0.f;
    for (int i = threadIdx.x; i < n; i += 256) s += partials[i];
    red[threadIdx.x] = s;
    __syncthreads();
    #pragma unroll
    for (int off = 128; off > 0; off >>= 1) {
        if (threadIdx.x < off) red[threadIdx.x] += red[threadIdx.x + off];
        __syncthreads();
    }
    if (threadIdx.x == 0) out[0] = red[0] * (1.0f / (float)BATCH);
}

extern "C" void kernel_launch(void* const* d_in, const int* in_sizes, int n_in,
                              void* d_out, int out_size, void* d_ws, size_t ws_size,
                              hipStream_t stream) {
    const int*   pos_u    = (const int*)d_in[0];
    const int*   pos_v    = (const int*)d_in[1];
    const int*   neg_v    = (const int*)d_in[2];
    const float* u_weight = (const float*)d_in[3];
    const float* v_weight = (const float*)d_in[4];
    float*       out      = (float*)d_out;
    float*       partials = (float*)d_ws;

    const int nblocks = BATCH / 16;  // 1024
    cbow_tile_kernel<<<nblocks, 128, 0, stream>>>(
        pos_u, pos_v, neg_v, u_weight, v_weight, partials);
    cbow_reduce_kernel<<<1, 256, 0, stream>>>(partials, nblocks, out);
}